// SpatialAttention_51926154608663
// MI455X (gfx1250) — compile-verified
//
#include <hip/hip_runtime.h>
#include <math.h>

#define B_    32
#define NOBJ  2048
#define DD    1024
#define NSPLIT 8

typedef float v2f __attribute__((ext_vector_type(2)));
typedef float v8f __attribute__((ext_vector_type(8)));

// ---------------------------------------------------------------------------
// OUT[M,N] = scale * (A[M,K] @ B') + bias, fp32 via V_WMMA_F32_16X16X4_F32.
// TRANSB=1: B' = W^T with W:[N,K]  (out[m,n] = sum_k A[m,k] * W[n,k])
// TRANSB=0: B' = W   with W:[K,N]  (out[m,n] = sum_k A[m,k] * W[k,n])
// One wave (32 lanes) owns one 16x16 output tile; grid sized so every wave
// has a valid tile => EXEC stays all-ones for WMMA.
// A-operand layout (16x4 f32): lane<16 holds K=k0,k0+1 of row M=lane;
// lane>=16 holds K=k0+2,k0+3 of row M=lane-16. B (4x16) mirrors it.
// Two independent accumulator chains (K unrolled by 8) so back-to-back
// v_wmma ops overlap in the XDL pipeline instead of serializing on acc.
// ---------------------------------------------------------------------------
template <int TRANSB>
__global__ void wmma_gemm_f32(const float* __restrict__ A,
                              const float* __restrict__ W,
                              const float* __restrict__ bias,
                              float* __restrict__ OUT,
                              int M, int N, int K, float scale)
{
    const int lane = threadIdx.x & 31;
    const int wave = threadIdx.x >> 5;
    const int tile = blockIdx.x * (blockDim.x >> 5) + wave;
    const int ntn  = N >> 4;
    const int mt   = tile / ntn;
    const int nt   = tile - mt * ntn;
    if (mt * 16 >= M) return;                 // whole-wave uniform

    const int m0   = mt << 4;
    const int n0   = nt << 4;
    const int half = lane >> 4;               // 0 | 1
    const int l15  = lane & 15;
    const int arow = m0 + l15;
    const int bcol = n0 + l15;
    const int koff = half << 1;               // 0 | 2

    const float* Ar = A + (size_t)arow * K;
    const float* Wc = W + (size_t)bcol * K;   // used when TRANSB

    v8f acc0 = {};
    v8f acc1 = {};
    for (int k0 = 0; k0 < K; k0 += 8) {
        const int ka = k0 + koff;
        const int kb = ka + 4;
        v2f a0, b0, a1, b1;
        a0.x = Ar[ka];     a0.y = Ar[ka + 1];
        a1.x = Ar[kb];     a1.y = Ar[kb + 1];
        if (TRANSB) {
            b0.x = Wc[ka]; b0.y = Wc[ka + 1];
            b1.x = Wc[kb]; b1.y = Wc[kb + 1];
        } else {
            b0.x = W[(size_t)ka * N + bcol];
            b0.y = W[(size_t)(ka + 1) * N + bcol];
            b1.x = W[(size_t)kb * N + bcol];
            b1.y = W[(size_t)(kb + 1) * N + bcol];
        }
        acc0 = __builtin_amdgcn_wmma_f32_16x16x4_f32(
                   false, a0, false, b0, (short)0, acc0, false, false);
        acc1 = __builtin_amdgcn_wmma_f32_16x16x4_f32(
                   false, a1, false, b1, (short)0, acc1, false, false);
    }
    v8f acc = acc0 + acc1;

    const float bb = bias ? bias[bcol] : 0.0f;
#pragma unroll
    for (int r = 0; r < 8; ++r) {
        const int row = m0 + r + (half << 3); // VGPR r: M=r (lo half), r+8 (hi)
        OUT[(size_t)row * N + bcol] = acc[r] * scale + bb;
    }
}

// g = alpha1*q1 + alpha2*q2  (tiny elementwise, 32K elems)
__global__ void combine_q(const float* __restrict__ q1, const float* __restrict__ q2,
                          const float* __restrict__ a1, const float* __restrict__ a2,
                          float* __restrict__ g, int n)
{
    int i = blockIdx.x * blockDim.x + threadIdx.x;
    if (i < n) g[i] = a1[0] * q1[i] + a2[0] * q2[i];
}

// scores[b,n] = dot(c[b,:], x[b,n,:])  — streaming mat-vec, 256 MB read.
// 8 waves per block, one row per wave, float4 (b128) loads, c[b] in LDS.
__global__ void scores_kernel(const float* __restrict__ x,
                              const float* __restrict__ c,
                              float* __restrict__ scores)
{
    __shared__ __align__(16) float cs[DD];
    const int b  = blockIdx.y;
    const int n0 = blockIdx.x * 8;
    for (int i = threadIdx.x; i < DD; i += blockDim.x) cs[i] = c[b * DD + i];
    __syncthreads();

    const int wave = threadIdx.x >> 5;
    const int lane = threadIdx.x & 31;
    const int n    = n0 + wave;
    const float4* row = (const float4*)(x + ((size_t)b * NOBJ + n) * DD);
    const float4* cv  = (const float4*)cs;

    float p = 0.0f;
#pragma unroll
    for (int i = 0; i < DD / 128; ++i) {      // 8 iterations
        float4 xv = row[lane + 32 * i];
        float4 cc = cv[lane + 32 * i];
        p += xv.x * cc.x + xv.y * cc.y + xv.z * cc.z + xv.w * cc.w;
    }
    for (int off = 16; off > 0; off >>= 1) p += __shfl_xor(p, off, 32);
    if (lane == 0) scores[(size_t)b * NOBJ + n] = p;
}

// In-place softmax over N=2048 per batch; one block per b.
__global__ void softmax_kernel(float* __restrict__ s)
{
    float* row = s + (size_t)blockIdx.x * NOBJ;
    __shared__ float red[8];
    const int tid = threadIdx.x, lane = tid & 31, wave = tid >> 5;
    const int nw = blockDim.x >> 5;

    float m = -INFINITY;
    for (int i = tid; i < NOBJ; i += blockDim.x) m = fmaxf(m, row[i]);
    for (int off = 16; off > 0; off >>= 1) m = fmaxf(m, __shfl_xor(m, off, 32));
    if (lane == 0) red[wave] = m;
    __syncthreads();
    float mall = red[0];
    for (int w = 1; w < nw; ++w) mall = fmaxf(mall, red[w]);
    __syncthreads();

    float ssum = 0.0f;
    for (int i = tid; i < NOBJ; i += blockDim.x) {
        float e = expf(row[i] - mall);
        row[i] = e;
        ssum += e;
    }
    for (int off = 16; off > 0; off >>= 1) ssum += __shfl_xor(ssum, off, 32);
    if (lane == 0) red[wave] = ssum;
    __syncthreads();
    float tot = 0.0f;
    for (int w = 0; w < nw; ++w) tot += red[w];
    const float inv = 1.0f / tot;
    for (int i = tid; i < NOBJ; i += blockDim.x) row[i] *= inv;
}

// upart[split,b,:] = sum_{n in split} w[b,n] * x[b,n,:]  — second 256 MB read.
// 256 threads * float4 = full D per block; weights chunk staged in LDS.
__global__ void wsum_kernel(const float* __restrict__ x,
                            const float* __restrict__ w,
                            float* __restrict__ upart)
{
    const int b     = blockIdx.y;
    const int split = blockIdx.x;
    const int nbeg  = split * (NOBJ / NSPLIT);
    __shared__ float ws[NOBJ / NSPLIT];
    for (int i = threadIdx.x; i < NOBJ / NSPLIT; i += blockDim.x)
        ws[i] = w[(size_t)b * NOBJ + nbeg + i];
    __syncthreads();

    const int tid = threadIdx.x;
    const float4* xrow = (const float4*)(x + ((size_t)b * NOBJ + nbeg) * DD) + tid;
    float4 acc = {0.0f, 0.0f, 0.0f, 0.0f};
    for (int n = 0; n < NOBJ / NSPLIT; ++n) {
        __builtin_prefetch(xrow + (n + 16) * (DD / 4), 0, 1); // global_prefetch_b8
        const float wn = ws[n];
        float4 xv = xrow[n * (DD / 4)];
        acc.x += wn * xv.x; acc.y += wn * xv.y;
        acc.z += wn * xv.z; acc.w += wn * xv.w;
    }
    float4* out = (float4*)(upart + ((size_t)split * B_ + b) * DD) + tid;
    *out = acc;
}

// u = sum over splits (deterministic tree, no float atomics)
__global__ void ureduce_kernel(const float* __restrict__ upart, float* __restrict__ u)
{
    int i = blockIdx.x * blockDim.x + threadIdx.x;
    if (i >= B_ * DD) return;
    float s = 0.0f;
#pragma unroll
    for (int k = 0; k < NSPLIT; ++k) s += upart[(size_t)k * B_ * DD + i];
    u[i] = s;
}

extern "C" void kernel_launch(void* const* d_in, const int* in_sizes, int n_in,
                              void* d_out, int out_size, void* d_ws, size_t ws_size,
                              hipStream_t stream)
{
    const float* obj  = (const float*)d_in[0];   // [B, N, D]
    const float* h    = (const float*)d_in[1];   // [2, B, D]
    const float* q1_w = (const float*)d_in[2];
    const float* q1_b = (const float*)d_in[3];
    const float* q2_w = (const float*)d_in[4];
    const float* q2_b = (const float*)d_in[5];
    const float* wk_w = (const float*)d_in[6];
    // wk_b (d_in[7]) contributes a per-batch constant to scores -> softmax-invariant, dropped.
    const float* wv_w = (const float*)d_in[8];
    const float* wv_b = (const float*)d_in[9];
    const float* a1   = (const float*)d_in[10];
    const float* a2   = (const float*)d_in[11];
    float* out = (float*)d_out;                  // [B, 1, D] = 32x1024

    float* ws     = (float*)d_ws;
    float* q1     = ws;                          // 32*1024
    float* q2     = q1 + B_ * DD;
    float* g      = q2 + B_ * DD;
    float* c      = g  + B_ * DD;
    float* scores = c  + B_ * DD;                // 32*2048
    float* upart  = scores + B_ * NOBJ;          // 8*32*1024
    float* u      = upart + NSPLIT * B_ * DD;

    const float inv_sqrt_d = 1.0f / 32.0f;       // 1/sqrt(1024)

    // 128 tiles of 16x16, 4 waves/block -> 32 blocks of 128 threads.
    wmma_gemm_f32<1><<<32, 128, 0, stream>>>(h,           q1_w, q1_b, q1, B_, DD, DD, 1.0f);
    wmma_gemm_f32<1><<<32, 128, 0, stream>>>(h + B_ * DD, q2_w, q2_b, q2, B_, DD, DD, 1.0f);
    combine_q<<<(B_ * DD + 255) / 256, 256, 0, stream>>>(q1, q2, a1, a2, g, B_ * DD);
    // c = inv_sqrt_d * (g @ wk_w)   (W in [K,N] orientation => TRANSB=0)
    wmma_gemm_f32<0><<<32, 128, 0, stream>>>(g, wk_w, nullptr, c, B_, DD, DD, inv_sqrt_d);

    scores_kernel<<<dim3(NOBJ / 8, B_), 256, 0, stream>>>(obj, c, scores);
    softmax_kernel<<<B_, 256, 0, stream>>>(scores);
    wsum_kernel<<<dim3(NSPLIT, B_), 256, 0, stream>>>(obj, scores, upart);
    ureduce_kernel<<<(B_ * DD + 255) / 256, 256, 0, stream>>>(upart, u);

    // out = u @ wv_w^T + wv_b
    wmma_gemm_f32<1><<<32, 128, 0, stream>>>(u, wv_w, wv_b, out, B_, DD, DD, 1.0f);
}